// SSMCompressedMirasTitans_61804579389940
// MI455X (gfx1250) — compile-verified
//
#include <hip/hip_runtime.h>
#include <math.h>

// ---------------------------------------------------------------------------
// SSM compressed-memory Miras/Titans step for MI455X (gfx1250, wave32, WMMA).
//
// Roofline plan:
//   * read_out = k @ W^T : 137 GFLOP fp32 -> v_wmma_f32_16x16x4_f32 tiles,
//     fed by the Tensor Data Mover (tensor_load_to_lds, TENSORcnt) with
//     double-buffered LDS so DMA overlaps matrix math.
//   * gradW never materialized: gradB = -(err^T (kC))/N + L2*(W C),
//                               gradC = -(k^T (err B))/N + L2*(W^T B).
//     -> four rank-16 streaming passes (~17us of HBM at 23.3 TB/s) instead of
//        a second 137-GFLOP GEMM.
//   * Gate MLP / LN / updates: tiny kernels.
// ---------------------------------------------------------------------------

typedef float v2f __attribute__((ext_vector_type(2)));
typedef float v8f __attribute__((ext_vector_type(8)));
typedef unsigned int u32;
typedef u32 u32x4 __attribute__((ext_vector_type(4)));
typedef int  i32x4 __attribute__((ext_vector_type(4)));
typedef int  i32x8 __attribute__((ext_vector_type(8)));

#define D_MODEL 2048
#define RANK    16
#define NROWS   16384
#define LRATE   1e-3f
#define L2REG   1e-4f
#define MOMENT  0.9f
#define WSCALE  0.1f
#define LN_EPS  1e-5f

#if defined(__has_builtin)
#  if __has_builtin(__builtin_amdgcn_tensor_load_to_lds)
#    define HAVE_TDM 1
#  else
#    define HAVE_TDM 0
#  endif
#else
#  define HAVE_TDM 0
#endif

__device__ __forceinline__ float wave_reduce_sum(float x) {
#pragma unroll
  for (int off = 16; off > 0; off >>= 1)
    x += __shfl_down(x, off, 32);
  return x;
}

// ---------------------------------------------------------------------------
// W[i][j] = 0.1*tanh(dot(B[i,:], C[j,:])) + (i==j)*D[i]
// ---------------------------------------------------------------------------
__global__ __launch_bounds__(256) void build_W_kernel(
    const float* __restrict__ Bm, const float* __restrict__ Cm,
    const float* __restrict__ Dv, float* __restrict__ W) {
  int idx = blockIdx.x * 256 + threadIdx.x;          // 0 .. 4M-1
  int i = idx >> 11;
  int j = idx & (D_MODEL - 1);
  const float4* br = (const float4*)(Bm + (size_t)i * RANK);
  const float4* cr = (const float4*)(Cm + (size_t)j * RANK);
  float acc = 0.f;
#pragma unroll
  for (int q = 0; q < 4; ++q) {
    float4 b4 = br[q]; float4 c4 = cr[q];
    acc += b4.x * c4.x + b4.y * c4.y + b4.z * c4.z + b4.w * c4.w;
  }
  float w = WSCALE * tanhf(acc);
  if (i == j) w += Dv[i];
  W[idx] = w;
}

// ---------------------------------------------------------------------------
// Column-sum of k (for k_mean), deterministic two-stage reduction.
// ---------------------------------------------------------------------------
#define CS_GROUPS 64
__global__ __launch_bounds__(256) void colsum_part_kernel(
    const float* __restrict__ k, float* __restrict__ part) {
  int c  = blockIdx.x * 256 + threadIdx.x;           // column
  int r0 = blockIdx.y * (NROWS / CS_GROUPS);         // 256 rows per group
  float s = 0.f;
  for (int n = 0; n < NROWS / CS_GROUPS; ++n)
    s += k[(size_t)(r0 + n) * D_MODEL + c];
  part[(size_t)blockIdx.y * D_MODEL + c] = s;
}

__global__ __launch_bounds__(256) void colsum_reduce_kernel(
    const float* __restrict__ part, float* __restrict__ ksum) {
  int c = blockIdx.x * 256 + threadIdx.x;
  float s = 0.f;
  for (int y = 0; y < CS_GROUPS; ++y) s += part[(size_t)y * D_MODEL + c];
  ksum[c] = s;
}

// ---------------------------------------------------------------------------
// O[n][r] = sum_d X[n][d] * S[d][r]     (one wave per row, rank-16 skinny)
// ---------------------------------------------------------------------------
__global__ __launch_bounds__(256) void rowdot16_kernel(
    const float* __restrict__ X, const float* __restrict__ S,
    float* __restrict__ O, int nrows) {
  int wave = (int)((blockIdx.x * 256 + threadIdx.x) >> 5);
  int lane = threadIdx.x & 31;
  if (wave >= nrows) return;
  float acc[RANK];
#pragma unroll
  for (int r = 0; r < RANK; ++r) acc[r] = 0.f;
  size_t base = (size_t)wave * D_MODEL;
  for (int d = lane; d < D_MODEL; d += 32) {
    float xv = X[base + d];
    const float* srow = S + (size_t)d * RANK;
#pragma unroll
    for (int r = 0; r < RANK; ++r) acc[r] = fmaf(xv, srow[r], acc[r]);
  }
#pragma unroll
  for (int r = 0; r < RANK; ++r) acc[r] = wave_reduce_sum(acc[r]);
  if (lane == 0) {
#pragma unroll
    for (int r = 0; r < RANK; ++r) O[(size_t)wave * RANK + r] = acc[r];
  }
}

// Same but X := (X - X2): used for errB = (v - read_out) @ B
__global__ __launch_bounds__(256) void rowdiffdot16_kernel(
    const float* __restrict__ X, const float* __restrict__ X2,
    const float* __restrict__ S, float* __restrict__ O, int nrows) {
  int wave = (int)((blockIdx.x * 256 + threadIdx.x) >> 5);
  int lane = threadIdx.x & 31;
  if (wave >= nrows) return;
  float acc[RANK];
#pragma unroll
  for (int r = 0; r < RANK; ++r) acc[r] = 0.f;
  size_t base = (size_t)wave * D_MODEL;
  for (int d = lane; d < D_MODEL; d += 32) {
    float xv = X[base + d] - X2[base + d];
    const float* srow = S + (size_t)d * RANK;
#pragma unroll
    for (int r = 0; r < RANK; ++r) acc[r] = fmaf(xv, srow[r], acc[r]);
  }
#pragma unroll
  for (int r = 0; r < RANK; ++r) acc[r] = wave_reduce_sum(acc[r]);
  if (lane == 0) {
#pragma unroll
    for (int r = 0; r < RANK; ++r) O[(size_t)wave * RANK + r] = acc[r];
  }
}

// ---------------------------------------------------------------------------
// TDM: DMA one 128-row x 32-dword fp32 tile (row stride D_MODEL) into LDS,
// padding +1 dword after every 32 dwords -> LDS row stride 33 (bank-safe).
// D# packing per CDNA5 ISA.
// ---------------------------------------------------------------------------
#if HAVE_TDM
__device__ __forceinline__ void tdm_load_tile(u32 lds_byte_addr,
                                              const float* gtile,
                                              u32 tensor_rows) {
  unsigned long long ga = (unsigned long long)(size_t)gtile;
  u32x4 g0;
  g0[0] = 1u;                                        // count=1 (valid), rest 0
  g0[1] = lds_byte_addr;                             // lds_addr
  g0[2] = (u32)(ga & 0xFFFFFFFFu);                   // global_addr[31:0]
  g0[3] = (u32)((ga >> 32) & 0x01FFFFFFu) | (2u << 30); // addr[56:32], type=2
  i32x8 g1;
  g1[0] = (int)((2u << 16)     // data_size = 4 bytes
              | (1u << 20)     // pad_enable
              | (4u << 22)     // pad_interval: 32 dwords
              | (0u << 25));   // pad_amount: 1 dword
  g1[1] = (int)((D_MODEL & 0xFFFFu) << 16);          // tensor_dim0[15:0]
  g1[2] = (int)(((D_MODEL >> 16) & 0xFFFFu)          // tensor_dim0[31:16]
              | ((tensor_rows & 0xFFFFu) << 16));    // tensor_dim1[15:0]
  g1[3] = (int)(((tensor_rows >> 16) & 0xFFFFu)      // tensor_dim1[31:16]
              | (32u << 16));                        // tile_dim0 = 32
  g1[4] = (int)128;                                  // tile_dim1 = 128 rows
  g1[5] = (int)D_MODEL;                              // tensor_dim0_stride lo
  g1[6] = 0;
  g1[7] = 0;
  i32x4 gz = {0, 0, 0, 0};
#if __clang_major__ >= 23
  i32x8 gp = {0, 0, 0, 0, 0, 0, 0, 0};
  __builtin_amdgcn_tensor_load_to_lds(g0, g1, gz, gz, gp, 0);
#else
  __builtin_amdgcn_tensor_load_to_lds(g0, g1, gz, gz, 0);
#endif
}
#endif

// ---------------------------------------------------------------------------
// Main GEMM: read_out[m][n] = sum_k k[m][kk] * W[n][kk]   (fp32 WMMA 16x16x4)
// Block 256 thr (8 waves), 128x128 tile, KC=32 chunks, double-buffered LDS
// filled by the Tensor Data Mover; wave 0 drives TDM + s_wait_tensorcnt.
// ---------------------------------------------------------------------------
#define GBM 128
#define GBN 128
#define GBK 32
#define LDSS 33

__global__ __launch_bounds__(256) void gemm_kwT_kernel(
    const float* __restrict__ A, const float* __restrict__ Wm,
    float* __restrict__ O) {
  __shared__ float As[2][GBM * LDSS];
  __shared__ float Bs[2][GBN * LDSS];

  const int tid  = threadIdx.x;
  const int wave = tid >> 5;
  const int lane = tid & 31;
  const int half = lane >> 4;       // 0: K pair {0,1}, 1: K pair {2,3}
  const int lr   = lane & 15;       // M / N index within fragment
  const int mw   = wave & 3;        // wave m-offset: mw*32
  const int nw   = wave >> 2;       // wave n-offset: nw*64
  const size_t m0 = (size_t)blockIdx.y * GBM;
  const size_t n0 = (size_t)blockIdx.x * GBN;

  v8f acc[2][4];
#pragma unroll
  for (int sm = 0; sm < 2; ++sm)
#pragma unroll
    for (int sn = 0; sn < 4; ++sn)
#pragma unroll
      for (int q = 0; q < 8; ++q) acc[sm][sn][q] = 0.f;

#if HAVE_TDM
  if (tid < 32) {   // wave 0 drives the DMA engine
    tdm_load_tile((u32)(size_t)&As[0][0], A  + m0 * D_MODEL, NROWS);
    tdm_load_tile((u32)(size_t)&Bs[0][0], Wm + n0 * D_MODEL, D_MODEL);
  }
#endif

  int buf = 0;
  for (int kc = 0; kc < D_MODEL; kc += GBK) {
#if HAVE_TDM
    if (tid < 32) {
      if (kc + GBK < D_MODEL) {
        tdm_load_tile((u32)(size_t)&As[buf ^ 1][0],
                      A  + m0 * D_MODEL + kc + GBK, NROWS);
        tdm_load_tile((u32)(size_t)&Bs[buf ^ 1][0],
                      Wm + n0 * D_MODEL + kc + GBK, D_MODEL);
        // in-order per wave: <=2 outstanding => current buffer complete
        __builtin_amdgcn_s_wait_tensorcnt(2);
      } else {
        __builtin_amdgcn_s_wait_tensorcnt(0);
      }
    }
#else
    // fallback: synchronous staging (VMEM -> VGPR -> ds_store)
#pragma unroll
    for (int j = 0; j < 4; ++j) {
      int idx = tid + j * 256;
      int row = idx >> 3;
      int c4  = (idx & 7) * 4;
      float4 av = *(const float4*)(A  + (m0 + row) * D_MODEL + kc + c4);
      float4 bv = *(const float4*)(Wm + (n0 + row) * D_MODEL + kc + c4);
      As[buf][row * LDSS + c4 + 0] = av.x; As[buf][row * LDSS + c4 + 1] = av.y;
      As[buf][row * LDSS + c4 + 2] = av.z; As[buf][row * LDSS + c4 + 3] = av.w;
      Bs[buf][row * LDSS + c4 + 0] = bv.x; Bs[buf][row * LDSS + c4 + 1] = bv.y;
      Bs[buf][row * LDSS + c4 + 2] = bv.z; Bs[buf][row * LDSS + c4 + 3] = bv.w;
    }
#endif
    __syncthreads();   // publish current buffer to all 8 waves

#pragma unroll
    for (int ks = 0; ks < GBK / 4; ++ks) {
      const int kb = ks * 4 + 2 * half;
      v2f afrag[2];
      v2f bfrag[4];
#pragma unroll
      for (int sm = 0; sm < 2; ++sm) {
        const float* p = &As[buf][(mw * 32 + sm * 16 + lr) * LDSS + kb];
        afrag[sm].x = p[0]; afrag[sm].y = p[1];
      }
#pragma unroll
      for (int sn = 0; sn < 4; ++sn) {
        const float* p = &Bs[buf][(nw * 64 + sn * 16 + lr) * LDSS + kb];
        bfrag[sn].x = p[0]; bfrag[sn].y = p[1];
      }
#pragma unroll
      for (int sm = 0; sm < 2; ++sm)
#pragma unroll
        for (int sn = 0; sn < 4; ++sn)
          acc[sm][sn] = __builtin_amdgcn_wmma_f32_16x16x4_f32(
              false, afrag[sm], false, bfrag[sn], (short)0, acc[sm][sn],
              false, false);
    }
    __syncthreads();   // all reads done before next TDM overwrites buf^1
    buf ^= 1;
  }

  // C/D layout: VGPR q -> M = q + 8*half, N = lr
#pragma unroll
  for (int sm = 0; sm < 2; ++sm) {
#pragma unroll
    for (int sn = 0; sn < 4; ++sn) {
      size_t col = n0 + nw * 64 + sn * 16 + lr;
#pragma unroll
      for (int q = 0; q < 8; ++q) {
        size_t rowm = m0 + mw * 32 + sm * 16 + q + 8 * half;
        O[rowm * D_MODEL + col] = acc[sm][sn][q];
      }
    }
  }
}

// ---------------------------------------------------------------------------
// T1[d][r] = sum_n (v-ro)[n][d]*kC[n][r];  T2[d][r] = sum_n k[n][d]*errB[n][r]
// Deterministic: per-block register accumulation -> partial buffers.
// ---------------------------------------------------------------------------
#define T12_BLOCKS 64
__global__ __launch_bounds__(1024) void t12_part_kernel(
    const float* __restrict__ k, const float* __restrict__ v,
    const float* __restrict__ ro, const float* __restrict__ kC,
    const float* __restrict__ errB,
    float* __restrict__ T1p, float* __restrict__ T2p) {
  const int t  = threadIdx.x;
  const int r  = t & 15;
  const int dg = t >> 4;            // 0..63, owns 32 consecutive d's
  const int d0 = dg * 32;
  float t1[32], t2[32];
#pragma unroll
  for (int q = 0; q < 32; ++q) { t1[q] = 0.f; t2[q] = 0.f; }

  const int rows = NROWS / T12_BLOCKS;   // 256
  const int n0 = blockIdx.x * rows;
  for (int n = n0; n < n0 + rows; ++n) {
    float kcr = kC[(size_t)n * RANK + r];
    float ebr = errB[(size_t)n * RANK + r];
    size_t base = (size_t)n * D_MODEL + d0;
#pragma unroll
    for (int q = 0; q < 32; ++q) {
      float ev = v[base + q] - ro[base + q];
      float kv = k[base + q];
      t1[q] = fmaf(ev, kcr, t1[q]);
      t2[q] = fmaf(kv, ebr, t2[q]);
    }
  }
  size_t pbase = (size_t)blockIdx.x * (D_MODEL * RANK);
#pragma unroll
  for (int q = 0; q < 32; ++q) {
    T1p[pbase + (size_t)(d0 + q) * RANK + r] = t1[q];
    T2p[pbase + (size_t)(d0 + q) * RANK + r] = t2[q];
  }
}

__global__ __launch_bounds__(256) void t12_reduce_kernel(
    const float* __restrict__ T1p, const float* __restrict__ T2p,
    float* __restrict__ T1, float* __restrict__ T2) {
  int e = blockIdx.x * 256 + threadIdx.x;     // 0 .. 32767
  float s1 = 0.f, s2 = 0.f;
  for (int p = 0; p < T12_BLOCKS; ++p) {
    s1 += T1p[(size_t)p * (D_MODEL * RANK) + e];
    s2 += T2p[(size_t)p * (D_MODEL * RANK) + e];
  }
  T1[e] = s1; T2[e] = s2;
}

// ---------------------------------------------------------------------------
// WTB[i][r] = sum_j W[j][i] * B[j][r]   (transpose access, 32-col strips)
// ---------------------------------------------------------------------------
__global__ __launch_bounds__(256) void wtb_kernel(
    const float* __restrict__ W, const float* __restrict__ Bm,
    float* __restrict__ WTB) {
  __shared__ float red[8][32][RANK];
  const int c = threadIdx.x & 31;
  const int s = threadIdx.x >> 5;
  const int i = blockIdx.x * 32 + c;
  float acc[RANK];
#pragma unroll
  for (int r = 0; r < RANK; ++r) acc[r] = 0.f;
  for (int j = s; j < D_MODEL; j += 8) {
    float wv = W[(size_t)j * D_MODEL + i];
    const float* br = Bm + (size_t)j * RANK;
#pragma unroll
    for (int r = 0; r < RANK; ++r) acc[r] = fmaf(wv, br[r], acc[r]);
  }
#pragma unroll
  for (int r = 0; r < RANK; ++r) red[s][c][r] = acc[r];
  __syncthreads();
  if (s == 0) {
#pragma unroll
    for (int r = 0; r < RANK; ++r) {
      float sum = acc[r];
      for (int q = 1; q < 8; ++q) sum += red[q][c][r];
      WTB[(size_t)i * RANK + r] = sum;
    }
  }
}

// ---------------------------------------------------------------------------
// LayerNorm of k_mean (single block)
// ---------------------------------------------------------------------------
__global__ __launch_bounds__(256) void ln_kernel(
    const float* __restrict__ ksum, const float* __restrict__ gamma,
    const float* __restrict__ beta, float* __restrict__ xn) {
  __shared__ float red[256];
  __shared__ float mu_s, inv_s;
  const int t = threadIdx.x;
  const float invN = 1.0f / (float)NROWS;

  float local = 0.f;
  for (int d = t; d < D_MODEL; d += 256) local += ksum[d] * invN;
  red[t] = local; __syncthreads();
  for (int s = 128; s > 0; s >>= 1) { if (t < s) red[t] += red[t + s]; __syncthreads(); }
  if (t == 0) mu_s = red[0] / (float)D_MODEL;
  __syncthreads();
  float mu = mu_s;

  float lv = 0.f;
  for (int d = t; d < D_MODEL; d += 256) {
    float x = ksum[d] * invN - mu;
    lv += x * x;
  }
  red[t] = lv; __syncthreads();
  for (int s = 128; s > 0; s >>= 1) { if (t < s) red[t] += red[t + s]; __syncthreads(); }
  if (t == 0) inv_s = rsqrtf(red[0] / (float)D_MODEL + LN_EPS);
  __syncthreads();
  float inv = inv_s;

  for (int d = t; d < D_MODEL; d += 256)
    xn[d] = (ksum[d] * invN - mu) * inv * gamma[d] + beta[d];
}

// h[i] = silu(dot(xn, W1[i,:]) + b1[i])   (one wave per row)
__global__ __launch_bounds__(256) void gemv_silu_kernel(
    const float* __restrict__ W1, const float* __restrict__ b1,
    const float* __restrict__ xn, float* __restrict__ h) {
  int row  = (int)((blockIdx.x * 256 + threadIdx.x) >> 5);
  int lane = threadIdx.x & 31;
  float acc = 0.f;
  size_t base = (size_t)row * D_MODEL;
  for (int d = lane; d < D_MODEL; d += 32) acc = fmaf(W1[base + d], xn[d], acc);
  acc = wave_reduce_sum(acc);
  if (lane == 0) {
    float z = acc + b1[row];
    h[row] = z / (1.f + expf(-z));
  }
}

// alpha = sigmoid(dot(h, W2) + b2)   (single block)
__global__ __launch_bounds__(256) void alpha_kernel(
    const float* __restrict__ h, const float* __restrict__ W2,
    const float* __restrict__ b2, float* __restrict__ alpha) {
  __shared__ float red[256];
  const int t = threadIdx.x;
  float local = 0.f;
  for (int d = t; d < D_MODEL; d += 256) local = fmaf(h[d], W2[d], local);
  red[t] = local; __syncthreads();
  for (int s = 128; s > 0; s >>= 1) { if (t < s) red[t] += red[t + s]; __syncthreads(); }
  if (t == 0) {
    float z = red[0] + b2[0];
    alpha[0] = 1.f / (1.f + expf(-z));
  }
}

// Momentum + retention update for B, C, S_B, S_C (+ alpha output)
__global__ __launch_bounds__(256) void final_update_kernel(
    const float* __restrict__ Bm, const float* __restrict__ Cm,
    const float* __restrict__ SB, const float* __restrict__ SC,
    const float* __restrict__ T1, const float* __restrict__ T2,
    const float* __restrict__ WC, const float* __restrict__ WTB,
    const float* __restrict__ alpha,
    float* __restrict__ oB, float* __restrict__ oC,
    float* __restrict__ oSB, float* __restrict__ oSC,
    float* __restrict__ oAl) {
  int e = blockIdx.x * 256 + threadIdx.x;     // 0 .. 32767
  float a = alpha[0];
  const float invN = 1.0f / ((float)NROWS + 1e-8f);
  float gB = -T1[e] * invN + L2REG * WC[e];
  float gC = -T2[e] * invN + L2REG * WTB[e];
  float sb = MOMENT * SB[e] - LRATE * gB;
  float sc = MOMENT * SC[e] - LRATE * gC;
  oSB[e] = sb; oSC[e] = sc;
  oB[e] = (1.f - a) * Bm[e] + sb;
  oC[e] = (1.f - a) * Cm[e] + sc;
  if (e == 0) oAl[0] = a;
}

// ---------------------------------------------------------------------------
extern "C" void kernel_launch(void* const* d_in, const int* in_sizes, int n_in,
                              void* d_out, int out_size, void* d_ws, size_t ws_size,
                              hipStream_t stream) {
  const float* k   = (const float*)d_in[0];
  const float* v   = (const float*)d_in[1];
  const float* Bm  = (const float*)d_in[2];
  const float* Cm  = (const float*)d_in[3];
  const float* Dv  = (const float*)d_in[4];
  const float* SB  = (const float*)d_in[5];
  const float* SC  = (const float*)d_in[6];
  const float* lng = (const float*)d_in[7];
  const float* lnb = (const float*)d_in[8];
  const float* W1  = (const float*)d_in[9];
  const float* b1  = (const float*)d_in[10];
  const float* W2  = (const float*)d_in[11];
  const float* b2  = (const float*)d_in[12];

  float* out = (float*)d_out;
  float* ro  = out;                                     // [16384 x 2048]
  float* oB  = out + (size_t)NROWS * D_MODEL;           // [2048 x 16]
  float* oC  = oB  + D_MODEL * RANK;
  float* oSB = oC  + D_MODEL * RANK;
  float* oSC = oSB + D_MODEL * RANK;
  float* oAl = oSC + D_MODEL * RANK;                    // scalar

  float* ws      = (float*)d_ws;
  float* W       = ws;                                            // 4,194,304
  float* kC      = W       + (size_t)D_MODEL * D_MODEL;           //   262,144
  float* errB    = kC      + (size_t)NROWS * RANK;                //   262,144
  float* colpart = errB    + (size_t)NROWS * RANK;                //   131,072
  float* ksum    = colpart + (size_t)CS_GROUPS * D_MODEL;         //     2,048
  float* T1p     = ksum    + D_MODEL;                             // 2,097,152
  float* T2p     = T1p     + (size_t)T12_BLOCKS * D_MODEL * RANK; // 2,097,152
  float* T1      = T2p     + (size_t)T12_BLOCKS * D_MODEL * RANK; //    32,768
  float* T2      = T1      + D_MODEL * RANK;
  float* WC      = T2      + D_MODEL * RANK;
  float* WTB     = WC      + D_MODEL * RANK;
  float* xn      = WTB     + D_MODEL * RANK;                      //     2,048
  float* hbuf    = xn      + D_MODEL;                             //     2,048
  float* alph    = hbuf    + D_MODEL;                             //         1

  // 1) memory matrix W = 0.1*tanh(B C^T) + diag(D)
  build_W_kernel<<<(D_MODEL * D_MODEL) / 256, 256, 0, stream>>>(Bm, Cm, Dv, W);

  // 2) column sums of k (for k_mean), and kC = k @ C
  colsum_part_kernel<<<dim3(D_MODEL / 256, CS_GROUPS), 256, 0, stream>>>(k, colpart);
  colsum_reduce_kernel<<<D_MODEL / 256, 256, 0, stream>>>(colpart, ksum);
  rowdot16_kernel<<<(NROWS * 32) / 256, 256, 0, stream>>>(k, Cm, kC, NROWS);

  // 3) read_out = k @ W^T   (WMMA fp32 16x16x4, TDM-fed, double-buffered)
  gemm_kwT_kernel<<<dim3(D_MODEL / GBN, NROWS / GBM), 256, 0, stream>>>(k, W, ro);

  // 4) errB = (v - read_out) @ B
  rowdiffdot16_kernel<<<(NROWS * 32) / 256, 256, 0, stream>>>(v, ro, Bm, errB, NROWS);

  // 5) T1 = err^T @ kC, T2 = k^T @ errB  (partials -> reduce, deterministic)
  t12_part_kernel<<<T12_BLOCKS, 1024, 0, stream>>>(k, v, ro, kC, errB, T1p, T2p);
  t12_reduce_kernel<<<(D_MODEL * RANK) / 256, 256, 0, stream>>>(T1p, T2p, T1, T2);

  // 6) WC = W @ C,  WTB = W^T @ B
  rowdot16_kernel<<<(D_MODEL * 32) / 256, 256, 0, stream>>>(W, Cm, WC, D_MODEL);
  wtb_kernel<<<D_MODEL / 32, 256, 0, stream>>>(W, Bm, WTB);

  // 7) retention gate: LN -> W1+SiLU -> W2 -> sigmoid
  ln_kernel<<<1, 256, 0, stream>>>(ksum, lng, lnb, xn);
  gemv_silu_kernel<<<(D_MODEL * 32) / 256, 256, 0, stream>>>(W1, b1, xn, hbuf);
  alpha_kernel<<<1, 256, 0, stream>>>(hbuf, W2, b2, alph);

  // 8) momentum + retention updates
  final_update_kernel<<<(D_MODEL * RANK) / 256, 256, 0, stream>>>(
      Bm, Cm, SB, SC, T1, T2, WC, WTB, alph, oB, oC, oSB, oSC, oAl);
}